// RieDFMLinkPredictionHead_40913858461829
// MI455X (gfx1250) — compile-verified
//
#include <hip/hip_runtime.h>
#include <math.h>

typedef float v2f __attribute__((ext_vector_type(2)));
typedef float v8f __attribute__((ext_vector_type(8)));

#define DTOT 128
#define NQ   64
#define GAMMA_C 12.0f
#define EPS_C   1e-6f
#define WS_FLOATS (NQ * DTOT + NQ)   // 8192 A-fragment floats + 64 tnorm floats

// LDS/ws A-fragment index for query element (row, col):
// frag (kt, m): kt = col/4, m = row/16, lane = (row&15) | (khi<<4), khi=(col&3)/2, j=col&1
__device__ __forceinline__ int aidx(int row, int col) {
    int kt  = col >> 2;
    int rem = col & 3;
    int khi = rem >> 1;
    int j   = rem & 1;
    int m   = row >> 4;
    int lane = (row & 15) | (khi << 4);
    return ((kt * 4 + m) * 32 + lane) * 2 + j;
}

// ---- translate one query row b, emitting A-fragment-layout elements via sink ----
template <typename Sink>
__device__ __forceinline__ void translate_row(int b,
                                              const int* __restrict__ h_ids,
                                              const int* __restrict__ r_ids,
                                              const float* __restrict__ ent,
                                              const float* __restrict__ rel,
                                              Sink&& emit, float& tnorm_out) {
    const float* __restrict__ h = ent + (size_t)h_ids[b] * DTOT;
    const float* __restrict__ r = rel + (size_t)r_ids[b] * DTOT;

    // hyperbolic (dims 0..63), Minkowski signature (-,+,...,+)
    float ip = -h[0] * r[0];
    for (int i = 1; i < 64; ++i) ip += h[i] * r[i];
    float mvv = 0.f;
    for (int i = 0; i < 64; ++i) {
        float v = r[i] + ip * h[i];
        mvv += (i == 0) ? -v * v : v * v;
    }
    float vn  = sqrtf(fmaxf(mvv, EPS_C));
    float chv = coshf(vn);
    float shv = sinhf(vn) / vn;
    for (int i = 0; i < 64; ++i) {
        float v = r[i] + ip * h[i];
        float t = chv * h[i] + shv * v;
        if (i == 0) t = -t;                 // fold sign vector into th
        emit(aidx(b, i), t);
    }
    // spherical (dims 64..95)
    float ips = 0.f;
    for (int i = 0; i < 32; ++i) ips += h[64 + i] * r[64 + i];
    float nv2 = 0.f;
    for (int i = 0; i < 32; ++i) {
        float v = r[64 + i] - ips * h[64 + i];
        nv2 += v * v;
    }
    float vns = fmaxf(sqrtf(nv2), EPS_C);
    float cv = cosf(vns);
    float sv = sinf(vns) / vns;
    for (int i = 0; i < 32; ++i) {
        float v = r[64 + i] - ips * h[64 + i];
        emit(aidx(b, 64 + i), cv * h[64 + i] + sv * v);
    }
    // euclidean (dims 96..127)
    float tn = 0.f;
    for (int i = 0; i < 32; ++i) {
        float t = h[96 + i] + r[96 + i];
        emit(aidx(b, 96 + i), t);
        tn += t * t;
    }
    tnorm_out = tn;
}

// ---- tiny one-block kernel: translate -> workspace (A-frag layout + tnorm) ----
__global__ void __launch_bounds__(NQ)
RieDFM_translate_kernel(const int* __restrict__ h_ids,
                        const int* __restrict__ r_ids,
                        const float* __restrict__ ent,
                        const float* __restrict__ rel,
                        float* __restrict__ ws) {
    const int b = threadIdx.x;
    float tn;
    auto sink = [&](int idx, float v) { ws[idx] = v; };
    translate_row(b, h_ids, r_ids, ent, rel, sink, tn);
    ws[NQ * DTOT + b] = tn;
}

// ---- main kernel: WMMA fp32 distance GEMM + transcendental epilogue ----
// FROM_WS=true : phase 1 is a coalesced ws->LDS copy (translate precomputed)
// FROM_WS=false: phase 1 recomputes translate per block (no-workspace fallback)
template <bool FROM_WS>
__global__ void __launch_bounds__(128)
RieDFM_head_kernel(const int* __restrict__ h_ids,
                   const int* __restrict__ r_ids,
                   const float* __restrict__ ent,
                   const float* __restrict__ rel,
                   const float* __restrict__ ws,
                   float* __restrict__ out,
                   int E) {
    __shared__ float Alds[NQ * DTOT];   // 32 KB: A fragments, WMMA layout
    __shared__ float tnorm_s[NQ];       // ||te||^2 per query row

    const int tid = threadIdx.x;

    // ---- Phase 1: stage query matrix into LDS ----
    if (FROM_WS) {
        const float4* __restrict__ src = (const float4*)ws;
        float4* dst = (float4*)Alds;
        #pragma unroll
        for (int i = 0; i < (NQ * DTOT / 4) / 128; ++i)   // 16 float4 per thread
            dst[tid + i * 128] = src[tid + i * 128];
        if (tid < NQ) tnorm_s[tid] = ws[NQ * DTOT + tid];
    } else {
        if (tid < NQ) {
            float tn;
            auto sink = [&](int idx, float v) { Alds[idx] = v; };
            translate_row(tid, h_ids, r_ids, ent, rel, sink, tn);
            tnorm_s[tid] = tn;
        }
    }
    __syncthreads();

    // ---- Phase 2: WMMA f32 GEMM, 16 entities per wave, M=64 ----
    const int wave = tid >> 5;
    const int lane = tid & 31;
    const int nlo  = lane & 15;       // entity column within tile
    const int khi  = lane >> 4;       // which K-pair this half-wave holds
    const int nb   = blockIdx.x * 64 + wave * 16;

    const int n  = nb + nlo;
    const bool valid = (n < E);
    const int nc = valid ? n : (E - 1);           // clamp loads, keep EXEC all-1 for WMMA
    const float* __restrict__ erow = ent + (size_t)nc * DTOT + 2 * khi;

    v8f acch[4], accs[4], acce[4];
    #pragma unroll
    for (int m = 0; m < 4; ++m) { acch[m] = (v8f)0.f; accs[m] = (v8f)0.f; acce[m] = (v8f)0.f; }

    float ep = 0.f;  // per-lane partial of ||ee||^2

    #pragma unroll
    for (int kt = 0; kt < 32; ++kt) {
        // B fragment 4x16 (K x N): lane holds ent[n][4kt + 2*khi + {0,1}]
        v2f bfrag = *(const v2f*)(erow + 4 * kt);
        if (kt >= 24) ep += bfrag.x * bfrag.x + bfrag.y * bfrag.y;

        const float* abase = &Alds[(kt * 4) * 64 + lane * 2];
        #pragma unroll
        for (int m = 0; m < 4; ++m) {
            v2f afrag = *(const v2f*)(abase + m * 64);
            if (kt < 16) {
                acch[m] = __builtin_amdgcn_wmma_f32_16x16x4_f32(
                    false, afrag, false, bfrag, (short)0, acch[m], false, false);
            } else if (kt < 24) {
                accs[m] = __builtin_amdgcn_wmma_f32_16x16x4_f32(
                    false, afrag, false, bfrag, (short)0, accs[m], false, false);
            } else {
                acce[m] = __builtin_amdgcn_wmma_f32_16x16x4_f32(
                    false, afrag, false, bfrag, (short)0, acce[m], false, false);
            }
        }
    }

    // complete ||ee||^2: half-wave swap (wave32)
    float en = ep + __shfl_xor(ep, 16, 32);

    // ---- Phase 3: epilogue ----
    #pragma unroll
    for (int m = 0; m < 4; ++m) {
        #pragma unroll
        for (int j = 0; j < 8; ++j) {
            int row = 16 * m + 8 * khi + j;      // C/D layout: lanes>=16 hold M=8..15
            float ih = acch[m][j];               // inner_h (sign already folded)
            float is = accs[m][j];
            float ie = acce[m][j];

            float x  = fmaxf(-ih, 1.0f + EPS_C);
            float dh = __logf(x + sqrtf(x * x - 1.0f));        // acosh via v_log_f32
            float cs = fminf(fmaxf(is, -1.0f + EPS_C), 1.0f - EPS_C);
            float dsv = acosf(cs);
            float de2 = fmaxf(tnorm_s[row] + en - 2.0f * ie, 0.0f);

            float dist = sqrtf(dh * dh + dsv * dsv + de2 + EPS_C);
            if (valid) out[(size_t)row * E + n] = GAMMA_C - dist;
        }
    }
}

extern "C" void kernel_launch(void* const* d_in, const int* in_sizes, int n_in,
                              void* d_out, int out_size, void* d_ws, size_t ws_size,
                              hipStream_t stream) {
    const int*   h_ids = (const int*)d_in[0];
    const int*   r_ids = (const int*)d_in[1];
    const float* ent   = (const float*)d_in[2];
    const float* rel   = (const float*)d_in[3];
    float*       out   = (float*)d_out;

    const int E = in_sizes[2] / DTOT;           // 200000
    const int nblocks = (E + 63) / 64;          // 64 entities per block (4 waves x 16)

    if (ws_size >= WS_FLOATS * sizeof(float)) {
        float* ws = (float*)d_ws;
        RieDFM_translate_kernel<<<1, NQ, 0, stream>>>(h_ids, r_ids, ent, rel, ws);
        RieDFM_head_kernel<true><<<nblocks, 128, 0, stream>>>(h_ids, r_ids, ent, rel,
                                                              ws, out, E);
    } else {
        RieDFM_head_kernel<false><<<nblocks, 128, 0, stream>>>(h_ids, r_ids, ent, rel,
                                                               nullptr, out, E);
    }
}